// ScaleInvariantDepthLoss_16183436771445
// MI455X (gfx1250) — compile-verified
//
#include <hip/hip_runtime.h>
#include <math.h>

// Problem constants (from reference setup_inputs)
#define BB 4
#define HH 192
#define WW 640
#define NN (HH * WW)            // 122880 elements per batch (T=1)
#define NITER 100
#define KK 50
#define CHUNK 4096              // floats per staged LDS chunk (16 KB)
#define NCHUNK (NN / CHUNK)     // 30, exact
#define CAND_MAX 1024
#define NPIX (BB * HH * WW)     // 491520
#define NB (NPIX / 256)         // 1920 loss blocks, exact

#define MIN_DV 0.01f
#define MAX_DV 50.0f
#define MAX_SCALE_V 100.0f

// ---- CDNA5 async global->LDS path (guarded; falls back to prefetch) ----
#if defined(__HIP_DEVICE_COMPILE__) && defined(__gfx1250__) && \
    __has_builtin(__builtin_amdgcn_global_load_async_to_lds_b128)
#define HAVE_ASYNC_LDS 1
#else
#define HAVE_ASYNC_LDS 0
#endif

#if HAVE_ASYNC_LDS
#define GAS __attribute__((address_space(1)))
#define LAS __attribute__((address_space(3)))
typedef int v4i __attribute__((vector_size(16)));
#if __has_builtin(__builtin_amdgcn_s_wait_asynccnt)
#define WAIT_ASYNC(n) __builtin_amdgcn_s_wait_asynccnt(n)
#else
#define WAIT_ASYNC(n) asm volatile("s_wait_asynccnt " #n)
#endif
#endif

// ---------------- block reductions (wave32-aware, deterministic) --------
__device__ __forceinline__ int blockReduceSumI(int v, int* red) {
  for (int o = 16; o > 0; o >>= 1) v += __shfl_xor(v, o, 32);
  const int wid = threadIdx.x >> 5, lane = threadIdx.x & 31;
  if (lane == 0) red[wid] = v;
  __syncthreads();
  const int nw = (blockDim.x + 31) >> 5;
  v = (threadIdx.x < nw) ? red[threadIdx.x] : 0;
  if (wid == 0) {
    for (int o = 16; o > 0; o >>= 1) v += __shfl_xor(v, o, 32);
    if (lane == 0) red[32] = v;
  }
  __syncthreads();
  v = red[32];
  __syncthreads();
  return v;
}

__device__ __forceinline__ float blockReduceSumF(float v, float* red) {
  for (int o = 16; o > 0; o >>= 1) v += __shfl_xor(v, o, 32);
  const int wid = threadIdx.x >> 5, lane = threadIdx.x & 31;
  if (lane == 0) red[wid] = v;
  __syncthreads();
  const int nw = (blockDim.x + 31) >> 5;
  v = (threadIdx.x < nw) ? red[threadIdx.x] : 0.0f;
  if (wid == 0) {
    for (int o = 16; o > 0; o >>= 1) v += __shfl_xor(v, o, 32);
    if (lane == 0) red[32] = v;
  }
  __syncthreads();
  v = red[32];
  __syncthreads();
  return v;
}

// ---------------- K1: ratio = valid ? gt/(aux+1e-8) : NaN ---------------
__global__ __launch_bounds__(256) void ratio_kernel(const float* __restrict__ gt,
                                                    const float* __restrict__ pred,
                                                    float* __restrict__ ratio) {
  const int i = blockIdx.x * 256 + threadIdx.x;
  if (i < BB * NN) {
    const float g = gt[i], p = pred[i];
    const bool v = (p > MIN_DV) && (g > MIN_DV) && (g < MAX_DV);
    ratio[i] = v ? (g / (p + 1e-8f)) : __builtin_nanf("");
  }
}

// ---- K2: per-(b,iter) top-50 / median candidate / inlier count ---------
// One 256-thread WG per (b, iter). Streams 480 KB of scores exactly once,
// double-buffered through LDS via async global->LDS loads on gfx1250.
__global__ __launch_bounds__(256) void ransac_kernel(const float* __restrict__ scores,
                                                     const float* __restrict__ ratio,
                                                     float* __restrict__ cand,
                                                     int* __restrict__ counts) {
  __shared__ float sbuf[2][CHUNK];
  __shared__ float candScore[CAND_MAX];
  __shared__ int candIdx[CAND_MAX];
  __shared__ int topIdx[KK];
  __shared__ float topRatio[KK];
  __shared__ int nCandSh;
  __shared__ float medSh;
  __shared__ int red[34];

  const int tid = threadIdx.x;
  const int b = blockIdx.x / NITER;
  const int it = blockIdx.x % NITER;
  const int bbase = b * NN;
  const float* sBase = scores + ((size_t)b * NITER + it) * (size_t)NN;

  // Threshold ladder: level 0 expects ~256 survivors among 122880 uniforms.
  const float levels[5] = {1.0f - 256.0f / (float)NN, 0.98f, 0.9f, 0.5f, -2.0f};
  int level = 0;
  for (;;) {
    if (tid == 0) nCandSh = 0;
    __syncthreads();
    const float T = levels[level];
    if (level == 0) {
#if HAVE_ASYNC_LDS
      GAS char* gp = (GAS char*)sBase;
      LAS char* lp0 = (LAS char*)&sbuf[0][0];
      LAS char* lp1 = (LAS char*)&sbuf[1][0];
      // prologue: chunk 0 -> buffer 0 (4 x b128 per thread)
      for (int j = 0; j < 4; ++j) {
        const int off = (j * 256 + tid) * 16;
        __builtin_amdgcn_global_load_async_to_lds_b128((GAS v4i*)(gp + off),
                                                       (LAS v4i*)(lp0 + off), 0, 0);
      }
      for (int c = 0; c < NCHUNK; ++c) {
        const int cur = c & 1;
        if (c + 1 < NCHUNK) {
          GAS char* gnext = gp + (size_t)(c + 1) * CHUNK * 4;
          LAS char* lnext = cur ? lp0 : lp1;
          for (int j = 0; j < 4; ++j) {
            const int off = (j * 256 + tid) * 16;
            __builtin_amdgcn_global_load_async_to_lds_b128((GAS v4i*)(gnext + off),
                                                           (LAS v4i*)(lnext + off), 0, 0);
          }
          WAIT_ASYNC(4);  // async loads complete in order: chunk c landed
        } else {
          WAIT_ASYNC(0);
        }
        __syncthreads();
        for (int k = 0; k < CHUNK / 256; ++k) {
          const int li = k * 256 + tid;
          const float v = sbuf[cur][li];
          if (v > T) {
            const int gidx = c * CHUNK + li;
            const float r = ratio[bbase + gidx];   // validity == !isnan (L2-hot)
            if (!__builtin_isnan(r)) {
              const int pos = atomicAdd(&nCandSh, 1);
              if (pos < CAND_MAX) { candScore[pos] = v; candIdx[pos] = gidx; }
            }
          }
        }
        __syncthreads();  // safe to overwrite sbuf[cur] with chunk c+2
      }
#else
      for (int i0 = tid; i0 < NN; i0 += 256) {
        if (i0 + 8192 < NN) __builtin_prefetch(sBase + i0 + 8192, 0, 0);
        const float v = sBase[i0];
        if (v > T) {
          const float r = ratio[bbase + i0];
          if (!__builtin_isnan(r)) {
            const int pos = atomicAdd(&nCandSh, 1);
            if (pos < CAND_MAX) { candScore[pos] = v; candIdx[pos] = i0; }
          }
        }
      }
#endif
    } else {
      // pathological fallback: rescan from global (L2-hot by now)
      for (int i0 = tid; i0 < NN; i0 += 256) {
        const float v = sBase[i0];
        if (v > T) {
          const float r = ratio[bbase + i0];
          if (!__builtin_isnan(r)) {
            const int pos = atomicAdd(&nCandSh, 1);
            if (pos < CAND_MAX) { candScore[pos] = v; candIdx[pos] = i0; }
          }
        }
      }
    }
    __syncthreads();
    const int nc = nCandSh;
    __syncthreads();
    if (nc >= KK || level == 4) break;
    ++level;
  }

  // Exact top-KK by (score desc, index asc) rank selection over candidates.
  if (tid == 0) medSh = __builtin_nanf("");
  if (tid < KK) topIdx[tid] = -1;
  __syncthreads();
  const int nC = (nCandSh < CAND_MAX) ? nCandSh : CAND_MAX;
  for (int j = tid; j < nC; j += 256) {
    const float s = candScore[j];
    const int id = candIdx[j];
    int rank = 0;
    for (int k = 0; k < nC; ++k) {
      const float sk = candScore[k];
      const int ik = candIdx[k];
      rank += (sk > s || (sk == s && ik < id)) ? 1 : 0;
    }
    if (rank < KK) topIdx[rank] = id;  // ranks are unique (total order)
  }
  __syncthreads();
  if (tid < KK) {
    const int id = topIdx[tid];
    topRatio[tid] = (id >= 0) ? ratio[bbase + id] : __builtin_nanf("");
  }
  __syncthreads();
  // Lower nanmedian of the 50 sampled ratios (rank method, NaN -> +inf).
  if (tid < KK) {
    const float v = topRatio[tid];
    const bool nn = !__builtin_isnan(v);
    const float key = nn ? v : __builtin_inff();
    int cnt = 0, rank = 0;
    for (int k = 0; k < KK; ++k) {
      const float u = topRatio[k];
      const bool un = !__builtin_isnan(u);
      cnt += un ? 1 : 0;
      const float ku = un ? u : __builtin_inff();
      rank += (ku < key || (ku == key && k < tid)) ? 1 : 0;
    }
    if (cnt > 0 && rank == (cnt - 1) / 2) medSh = v;
  }
  __syncthreads();
  const float c = medSh;

  // Inlier count over the (L2-resident) ratio array.
  int cnt = 0;
  for (int i = tid; i < NN; i += 256) {
    const float r = ratio[bbase + i];
    cnt += (fabsf(r - c) < 0.1f * (c + 1e-8f)) ? 1 : 0;  // NaN-safe: false
  }
  cnt = blockReduceSumI(cnt, red);
  if (tid == 0) {
    counts[b * NITER + it] = cnt;   // NaN cand -> 0 automatically
    cand[b * NITER + it] = c;
  }
}

// ---- K3: per-batch argmax + exact bit-bisection medians -> scale -------
__global__ __launch_bounds__(1024) void finalize_kernel(const float* __restrict__ ratio,
                                                        const float* __restrict__ cand,
                                                        const int* __restrict__ counts,
                                                        float* __restrict__ scaleArr) {
  __shared__ int red[34];
  __shared__ float cBestSh;
  const int b = blockIdx.x;
  const int base = b * NN;
  const int tid = threadIdx.x;

  int cv = 0;
  for (int i = tid; i < NN; i += blockDim.x)
    cv += __builtin_isnan(ratio[base + i]) ? 0 : 1;
  const int cntValid = blockReduceSumI(cv, red);

  // fallback = lower nanmedian of ratio: exact k-th smallest via bit bisection
  float fallback = __builtin_nanf("");
  if (cntValid > 0) {
    const int kth = (cntValid - 1) / 2 + 1;
    unsigned lo = 0u, hi = 0x7F800000u;  // ratios are positive finite
    while (lo < hi) {
      const unsigned mid = lo + ((hi - lo) >> 1);
      int cnt = 0;
      for (int i = tid; i < NN; i += blockDim.x) {
        const float x = ratio[base + i];
        cnt += (!__builtin_isnan(x) && (__float_as_uint(x) <= mid)) ? 1 : 0;
      }
      cnt = blockReduceSumI(cnt, red);
      if (cnt >= kth) hi = mid; else lo = mid + 1;
    }
    fallback = __uint_as_float(lo);
  }

  if (tid == 0) {
    int bi = 0, bc = counts[b * NITER];
    for (int i = 1; i < NITER; ++i) {
      const int ci = counts[b * NITER + i];
      if (ci > bc) { bc = ci; bi = i; }  // first-index tie-break (argmax)
    }
    cBestSh = cand[b * NITER + bi];
  }
  __syncthreads();
  const float c = cBestSh;

  int ciL = 0;
  for (int i = tid; i < NN; i += blockDim.x) {
    const float x = ratio[base + i];
    ciL += (fabsf(x - c) < 0.1f * (c + 1e-8f)) ? 1 : 0;
  }
  const int cntIn = blockReduceSumI(ciL, red);

  float med = __builtin_nanf("");
  if (cntIn > 0) {
    const int kth = (cntIn - 1) / 2 + 1;
    unsigned lo = 0u, hi = 0x7F800000u;
    while (lo < hi) {
      const unsigned mid = lo + ((hi - lo) >> 1);
      int cnt = 0;
      for (int i = tid; i < NN; i += blockDim.x) {
        const float x = ratio[base + i];
        const bool m = (fabsf(x - c) < 0.1f * (c + 1e-8f));
        cnt += (m && (__float_as_uint(x) <= mid)) ? 1 : 0;
      }
      cnt = blockReduceSumI(cnt, red);
      if (cnt >= kth) hi = mid; else lo = mid + 1;
    }
    med = __uint_as_float(lo);
  }

  if (tid == 0) {
    float best = (cntValid >= KK) ? med : fallback;
    if (__builtin_isnan(best)) best = fallback;
    if (__builtin_isnan(best)) best = 1.0f;
    best = fminf(fmaxf(best, 1.0f / MAX_SCALE_V), MAX_SCALE_V);
    scaleArr[b] = best;
  }
}

// ---- K4: loss partials (L1 + gradient terms), deterministic -----------
__global__ __launch_bounds__(256) void loss_kernel(const float* __restrict__ train,
                                                   const float* __restrict__ aux,
                                                   const float* __restrict__ scaleArr,
                                                   float* __restrict__ partial) {
  __shared__ float redf[34];
  const int i = blockIdx.x * 256 + threadIdx.x;
  float l1 = 0.f, msum = 0.f, ldx = 0.f, ndx = 0.f, ldy = 0.f, ndy = 0.f;
  if (i < NPIX) {
    const int b = i / (HH * WW);
    const int rem = i - b * (HH * WW);
    const int y = rem / WW;
    const int x = rem - y * WW;
    const float s = scaleArr[b];
    const float p = train[i];
    const float g = aux[i] * s;
    const float m = (p > MIN_DV && g > MIN_DV && g < MAX_DV) ? 1.f : 0.f;
    l1 = fabsf(p - g) * m;
    msum = m;
    if (x < WW - 1) {
      const float p1 = train[i + 1];
      const float g1 = aux[i + 1] * s;
      const float m1 = (p1 > MIN_DV && g1 > MIN_DV && g1 < MAX_DV) ? 1.f : 0.f;
      const float md = m * m1;
      ldx = fabsf((p1 - p) - (g1 - g)) * md;
      ndx = md;
    }
    if (y < HH - 1) {
      const float p2 = train[i + WW];
      const float g2 = aux[i + WW] * s;
      const float m2 = (p2 > MIN_DV && g2 > MIN_DV && g2 < MAX_DV) ? 1.f : 0.f;
      const float md = m * m2;
      ldy = fabsf((p2 - p) - (g2 - g)) * md;
      ndy = md;
    }
  }
  l1 = blockReduceSumF(l1, redf);
  msum = blockReduceSumF(msum, redf);
  ldx = blockReduceSumF(ldx, redf);
  ndx = blockReduceSumF(ndx, redf);
  ldy = blockReduceSumF(ldy, redf);
  ndy = blockReduceSumF(ndy, redf);
  if (threadIdx.x == 0) {
    partial[0 * NB + blockIdx.x] = l1;
    partial[1 * NB + blockIdx.x] = msum;
    partial[2 * NB + blockIdx.x] = ldx;
    partial[3 * NB + blockIdx.x] = ndx;
    partial[4 * NB + blockIdx.x] = ldy;
    partial[5 * NB + blockIdx.x] = ndy;
  }
}

// ---- K5: fixed-order final reduction + loss combine -------------------
__global__ __launch_bounds__(256) void reduce_final_kernel(const float* __restrict__ partial,
                                                           float* __restrict__ out) {
  __shared__ float redf[34];
  __shared__ float tot[6];
  for (int j = 0; j < 6; ++j) {
    float s = 0.f;
    for (int i = threadIdx.x; i < NB; i += 256) s += partial[j * NB + i];
    s = blockReduceSumF(s, redf);
    if (threadIdx.x == 0) tot[j] = s;
    __syncthreads();
  }
  if (threadIdx.x == 0) {
    const float n = fmaxf(tot[1], 1.0f);
    out[0] = tot[0] / n + tot[2] / fmaxf(tot[3], 1e-8f) + tot[4] / fmaxf(tot[5], 1e-8f);
  }
}

extern "C" void kernel_launch(void* const* d_in, const int* in_sizes, int n_in,
                              void* d_out, int out_size, void* d_ws, size_t ws_size,
                              hipStream_t stream) {
  (void)in_sizes; (void)n_in; (void)out_size; (void)ws_size;
  const float* train  = (const float*)d_in[0];
  const float* aux    = (const float*)d_in[1];
  const float* gt     = (const float*)d_in[2];
  const float* scores = (const float*)d_in[3];
  float* out = (float*)d_out;

  // workspace layout (~2.0 MB total)
  float* ratio    = (float*)d_ws;                 // BB*NN
  float* cand     = ratio + (size_t)BB * NN;      // BB*NITER
  int*   counts   = (int*)(cand + BB * NITER);    // BB*NITER
  float* scaleArr = (float*)(counts + BB * NITER);// BB
  float* partial  = scaleArr + BB;                // 6*NB

  ratio_kernel<<<(BB * NN + 255) / 256, 256, 0, stream>>>(gt, aux, ratio);
  ransac_kernel<<<BB * NITER, 256, 0, stream>>>(scores, ratio, cand, counts);
  finalize_kernel<<<BB, 1024, 0, stream>>>(ratio, cand, counts, scaleArr);
  loss_kernel<<<NB, 256, 0, stream>>>(train, aux, scaleArr, partial);
  reduce_final_kernel<<<1, 256, 0, stream>>>(partial, out);
}